// MultiHeadTargetAttention_71588514889824
// MI455X (gfx1250) — compile-verified
//
#include <hip/hip_runtime.h>
#include <cmath>

// ---------------------------------------------------------------------------
// MultiHeadTargetAttention, algebraically fused for MI455X (gfx1250, wave32).
//
//   Mq = (W_q @ W_k^T) / sqrt(128)        [64x64]   (tiny, fp32 WMMA)
//   P  =  W_v @ W_o                       [64x64]   (tiny, fp32 WMMA)
//   Tt = target @ Mq                      [B x 64]  (dense fp32 WMMA GEMM)
//   s[b,l]  = Tt[b] . hist[b,l]           (streamer, memory-bound)
//   a[b,l]  = relu(s[b,l] * mask[b,l])
//   Ht[b]   = sum_l a[b,l] * hist[b,l]    [B x 64]
//   out     = Ht @ P + target             [B x 64]  (dense fp32 WMMA GEMM)
//
// Exact fp32 reassociation of the reference: K/V are never materialized.
// History stream (420 MB) bounds runtime at ~18.5 us on 23.3 TB/s HBM.
// Workspace use: 2*64*64 + 2*B*64 floats (~4.2 MB for B=8192).
// ---------------------------------------------------------------------------

typedef float v2f __attribute__((ext_vector_type(2)));
typedef float v8f __attribute__((ext_vector_type(8)));

#define D_IN   64
#define D_ATTN 128

// fp32 WMMA 16x16x4: A lane m=lane&15, pair K = k0 + 2*(lane>>4) + {0,1};
// B lane n=lane&15, same K pairing; D lane n=lane&15, VGPR r -> m = r + 8*(lane>>4).
__device__ __forceinline__ v8f wmma_f32_16x16x4(v2f a, v2f b, v8f c) {
#if __has_builtin(__builtin_amdgcn_wmma_f32_16x16x4_f32)
  return __builtin_amdgcn_wmma_f32_16x16x4_f32(false, a, false, b, (short)0, c,
                                               false, false);
#else
  // Marker fallback (histogram will show wmma==0 if the builtin is absent).
  c[0] += a[0] * b[0];
  return c;
#endif
}

// Kernel 1: one workgroup, 32 waves. Waves 0..15 -> 16x16 tiles of Mq,
// waves 16..31 -> tiles of P. K = 128 -> 32 chained WMMA f32 steps.
__global__ void prep_weights_kernel(const float* __restrict__ Wq,  // [64 x 128]
                                    const float* __restrict__ Wk,  // [64 x 128]
                                    const float* __restrict__ Wv,  // [64 x 128]
                                    const float* __restrict__ Wo,  // [128 x 64]
                                    float* __restrict__ Mq,        // [64 x 64]
                                    float* __restrict__ P,         // [64 x 64]
                                    float invScale) {
  const int lane = threadIdx.x & 31;
  const int wave = threadIdx.x >> 5;      // 0..31
  const bool doP = wave >= 16;
  const int t    = wave & 15;
  const int ti   = t >> 2;                // 16-row tile (0..3)
  const int tj   = t & 3;                 // 16-col tile (0..3)
  const int m    = lane & 15;
  const int half = lane >> 4;
  const int row  = ti * 16 + m;
  const int n    = tj * 16 + m;

  const float* Abase = doP ? Wv : Wq;     // A[row][k], row-major [64 x 128]
  const float* Bbase = doP ? Wo : Wk;     // B[k][n]: Wo row-major, Wk transposed
  const int bKs = doP ? D_IN : 1;         // stride of B over k
  const int bNs = doP ? 1 : D_ATTN;       // stride of B over n

  v8f acc = {};
  for (int k0 = 0; k0 < D_ATTN; k0 += 4) {
    const int k = k0 + 2 * half;
    v2f a = { Abase[row * D_ATTN + k], Abase[row * D_ATTN + k + 1] };
    v2f b = { Bbase[k * bKs + n * bNs], Bbase[(k + 1) * bKs + n * bNs] };
    acc = wmma_f32_16x16x4(a, b, acc);
  }

  float* dst = doP ? P : Mq;
  const float sc = doP ? 1.0f : invScale;
#pragma unroll
  for (int r = 0; r < 8; ++r)
    dst[(ti * 16 + half * 8 + r) * D_IN + tj * 16 + m] = acc[r] * sc;
}

// Kernels 2 & 4: dense [B x 64] @ [64 x 64] fp32 WMMA GEMM, one 16x16 tile
// per wave (4 waves/block cover the 64 output columns), 16 K-steps.
template <bool RESIDUAL>
__global__ void gemm64_kernel(const float* __restrict__ A,   // [B x 64]
                              const float* __restrict__ W,   // [64 x 64]
                              const float* __restrict__ R,   // [B x 64] or null
                              float* __restrict__ O) {       // [B x 64]
  const int lane = threadIdx.x & 31;
  const int wave = threadIdx.x >> 5;      // column tile 0..3
  const int m    = lane & 15;
  const int half = lane >> 4;
  const size_t row0 = (size_t)blockIdx.x * 16;
  const float* Arow = A + (row0 + m) * D_IN;
  const int n = wave * 16 + m;

  v8f acc = {};
#pragma unroll
  for (int k0 = 0; k0 < D_IN; k0 += 4) {
    const int k = k0 + 2 * half;
    v2f a = { Arow[k], Arow[k + 1] };
    v2f b = { W[k * D_IN + n], W[(k + 1) * D_IN + n] };
    acc = wmma_f32_16x16x4(a, b, acc);
  }

#pragma unroll
  for (int r = 0; r < 8; ++r) {
    const size_t orow = row0 + half * 8 + r;
    float v = acc[r];
    if (RESIDUAL) v += R[orow * D_IN + n];
    O[orow * D_IN + n] = v;
  }
}

// Kernel 3: the memory-bound streamer. One workgroup (8 waves) per batch.
// Each wave iteration: one global_load_b128 per lane = 2 history rows per
// wave (512 contiguous bytes), 16-lane butterfly dot reduction, mask+relu,
// FMA accumulate of h~ in VGPRs. Cross-wave reduce in LDS. L assumed even.
__global__ void attn_stream_kernel(const float* __restrict__ hist, // [B,L,64]
                                   const int*  __restrict__ mask,  // [B,L]
                                   const float* __restrict__ Tt,   // [B,64]
                                   float* __restrict__ Ht,         // [B,64]
                                   int L) {
  const int b    = blockIdx.x;
  const int lane = threadIdx.x & 31;
  const int wave = threadIdx.x >> 5;      // 0..7
  const int c    = lane & 15;             // which float4 of the 64-wide row
  const int hi   = lane >> 4;             // which row of the loaded pair

  const float4* h4 = (const float4*)(hist + (size_t)b * L * D_IN);
  const float4  t  = ((const float4*)(Tt + (size_t)b * D_IN))[c];
  const int* mrow  = mask + (size_t)b * L;

  float4 acc = make_float4(0.f, 0.f, 0.f, 0.f);
  const int npair = L >> 1;
  for (int pair = wave; pair < npair; pair += 8) {
    const int row = pair * 2 + hi;
    const float4 h = h4[row * 16 + c];
    float p = h.x * t.x + h.y * t.y + h.z * t.z + h.w * t.w;
    p += __shfl_xor(p, 1);
    p += __shfl_xor(p, 2);
    p += __shfl_xor(p, 4);
    p += __shfl_xor(p, 8);                // sums the 16 lanes of this row
    float s = p * (float)mrow[row];
    s = fmaxf(s, 0.0f);
    acc.x += s * h.x; acc.y += s * h.y; acc.z += s * h.z; acc.w += s * h.w;
  }

  __shared__ float red[16][D_IN];         // 4 KB
  float* r = &red[wave * 2 + hi][c * 4];
  r[0] = acc.x; r[1] = acc.y; r[2] = acc.z; r[3] = acc.w;
  __syncthreads();

  if (threadIdx.x < D_IN) {
    float s = 0.f;
#pragma unroll
    for (int g = 0; g < 16; ++g) s += red[g][threadIdx.x];
    Ht[(size_t)b * D_IN + threadIdx.x] = s;
  }
}

extern "C" void kernel_launch(void* const* d_in, const int* in_sizes, int n_in,
                              void* d_out, int out_size, void* d_ws,
                              size_t ws_size, hipStream_t stream) {
  const float* target = (const float*)d_in[0];  // [B, 64]
  const float* hist   = (const float*)d_in[1];  // [B, L, 64]
  const int*   mask   = (const int*)d_in[2];    // [B, 1, L]
  const float* Wq     = (const float*)d_in[3];  // [64, 128]
  const float* Wk     = (const float*)d_in[4];  // [64, 128]
  const float* Wv     = (const float*)d_in[5];  // [64, 128]
  const float* Wo     = (const float*)d_in[6];  // [128, 64]
  float* out = (float*)d_out;                   // [B, 64]

  const int B = in_sizes[0] / D_IN;             // 8192
  const int L = in_sizes[2] / B;                // 200

  float* ws = (float*)d_ws;
  float* Mq = ws;                               // 64*64
  float* P  = ws + 64 * 64;                     // 64*64
  float* Tt = ws + 2 * 64 * 64;                 // B*64
  float* Ht = Tt + (size_t)B * D_IN;            // B*64

  const float invScale = 1.0f / sqrtf((float)D_ATTN);

  prep_weights_kernel<<<1, 1024, 0, stream>>>(Wq, Wk, Wv, Wo, Mq, P, invScale);
  gemm64_kernel<false><<<B / 16, 128, 0, stream>>>(target, Mq, nullptr, Tt);
  attn_stream_kernel<<<B, 256, 0, stream>>>(hist, mask, Tt, Ht, L);
  gemm64_kernel<true><<<B / 16, 128, 0, stream>>>(Ht, P, target, out);
}